// AdaptiveStateAllocator_64424509440484
// MI455X (gfx1250) — compile-verified
//
#include <hip/hip_runtime.h>
#include <hip/hip_bf16.h>

typedef __attribute__((ext_vector_type(2))) float v2f;
typedef __attribute__((ext_vector_type(4))) float v4f;
typedef __attribute__((ext_vector_type(8))) float v8f;

#define B_DIM 32
#define S_DIM 2048
#define DIN 1024
#define DST 512
#define NSTATES 64
#define SCHUNK 32
#define SPER (S_DIM / SCHUNK) // 64

// ---------------------------------------------------------------------------
// Stage 1: split-S partial mean-pool. x[b, sc*SPER .. , :] summed into
// partial[b][sc][0..1023]. float4 loads, fully coalesced. 1024 blocks.
// ---------------------------------------------------------------------------
__global__ __launch_bounds__(256) void pool_stage1(const float* __restrict__ x,
                                                   float* __restrict__ partial) {
  const int sc = blockIdx.x;  // 0..31
  const int b  = blockIdx.y;  // 0..31
  const int t  = threadIdx.x; // 0..255 -> 256 float4 = 1024 floats per row
  const v4f* xp = (const v4f*)(x + ((size_t)b * S_DIM + (size_t)sc * SPER) * DIN);
  v4f acc = {0.f, 0.f, 0.f, 0.f};
#pragma unroll 4
  for (int s = 0; s < SPER; ++s)
    acc += xp[(size_t)s * (DIN / 4) + t];
  ((v4f*)partial)[((size_t)b * SCHUNK + sc) * (DIN / 4) + t] = acc;
}

// ---------------------------------------------------------------------------
// Stage 2: reduce partials over SCHUNK, scale by 1/S -> x_pooled[32][1024].
// ---------------------------------------------------------------------------
__global__ __launch_bounds__(256) void pool_stage2(const float* __restrict__ partial,
                                                   float* __restrict__ xpool) {
  const int d = blockIdx.x * 256 + threadIdx.x; // 0..1023
  const int b = blockIdx.y;
  float acc = 0.f;
#pragma unroll
  for (int sc = 0; sc < SCHUNK; ++sc)
    acc += partial[((size_t)b * SCHUNK + sc) * DIN + d];
  xpool[b * DIN + d] = acc * (1.0f / (float)S_DIM);
}

// ---------------------------------------------------------------------------
// out = relu(A @ W^T + bias), A:[M,K] row-major, W:[N,K] row-major.
// One wave per 16x16 output tile, K consumed 4 at a time by
// V_WMMA_F32_16X16X4_F32 (fp32 in, fp32 acc -> matches reference numerics).
// A fragment (ISA 7.12.2, 32-bit A 16x4): lanes 0-15 hold K=0,1; lanes 16-31
// hold K=2,3 for row M=lane%16. B fragment mirrors with N=lane%16.
// C/D: VGPR j = row m0+j+8*(lane>=16), col n0+lane%16.
// ---------------------------------------------------------------------------
__global__ __launch_bounds__(32) void wmma_gemm_relu(const float* __restrict__ A,
                                                     const float* __restrict__ W,
                                                     const float* __restrict__ bias,
                                                     float* __restrict__ out,
                                                     int M, int N, int K) {
  const int n0   = blockIdx.x * 16;
  const int m0   = blockIdx.y * 16;
  const int lane = threadIdx.x;      // 0..31, wave32, EXEC all ones
  const int half = lane >> 4;        // 0 | 1
  const int l    = lane & 15;
  const float* arow = A + (size_t)(m0 + l) * K + 2 * half;
  const float* wrow = W + (size_t)(n0 + l) * K + 2 * half;
  v8f c = {};
#pragma unroll 4
  for (int k0 = 0; k0 < K; k0 += 4) {
    v2f a = *(const v2f*)(arow + k0);   // {A[m,k0+2h], A[m,k0+2h+1]}
    v2f b = *(const v2f*)(wrow + k0);   // {W[n,k0+2h], W[n,k0+2h+1]}
    c = __builtin_amdgcn_wmma_f32_16x16x4_f32(
        /*neg_a=*/false, a, /*neg_b=*/false, b,
        /*c_mod=*/(short)0, c, /*reuse_a=*/false, /*reuse_b=*/false);
  }
  const float bn = bias[n0 + l];
#pragma unroll
  for (int j = 0; j < 8; ++j) {
    const int row = m0 + j + 8 * half;
    float v = c[j] + bn;
    out[(size_t)row * N + n0 + l] = v > 0.f ? v : 0.f;
  }
}

// ---------------------------------------------------------------------------
// Heads: complexity sigmoid -> num_states (round-half-even like jnp.round),
// importance logits -> stable descending ranks. Softmax/temperature skipped:
// temp = max(|t|,0.1) > 0 is monotonic, so top-k ranks depend only on logit
// order. rank[i] = #{j: L[j]>L[i]} + #{j<i: L[j]==L[i]} == argsort(argsort(-L)).
// ---------------------------------------------------------------------------
__global__ __launch_bounds__(64) void heads_kernel(const float* __restrict__ h2,
                                                   const float* __restrict__ W3,
                                                   const float* __restrict__ b3,
                                                   const float* __restrict__ hs,
                                                   const float* __restrict__ Ws2,
                                                   const float* __restrict__ bs2,
                                                   int* __restrict__ num_states,
                                                   int* __restrict__ ranks) {
  __shared__ float logits[NSTATES];
  const int t = threadIdx.x; // 0..63
  if (t < B_DIM) {
    float acc = 0.f;
    for (int k = 0; k < DST / 2; ++k) acc += h2[t * (DST / 2) + k] * W3[k];
    acc += b3[0];
    const float comp = 1.0f / (1.0f + expf(-acc));
    int n = (int)rintf(4.0f + comp * 60.0f);
    n = n < 4 ? 4 : (n > 64 ? 64 : n);
    num_states[t] = n;
  }
  {
    float acc = 0.f;
    for (int k = 0; k < DST / 2; ++k) acc += hs[t * (DST / 2) + k] * Ws2[k];
    logits[t] = acc + bs2[0];
  }
  __syncthreads();
  const float li = logits[t];
  int r = 0;
#pragma unroll
  for (int j = 0; j < NSTATES; ++j) {
    const float lj = logits[j];
    r += (lj > li || (lj == li && j < t)) ? 1 : 0;
  }
  ranks[t] = r;
}

// ---------------------------------------------------------------------------
// Broadcast state_bank -> allocated_states [32,64,512] (float4 stores, source
// is L2-resident 128KB) and emit state_mask [32,64] as 0.0/1.0 floats.
// ---------------------------------------------------------------------------
__global__ __launch_bounds__(256) void output_kernel(const float* __restrict__ bank,
                                                     const int* __restrict__ num_states,
                                                     const int* __restrict__ ranks,
                                                     float* __restrict__ out) {
  const int tid = blockIdx.x * 256 + threadIdx.x;
  const int NV4 = B_DIM * NSTATES * (DST / 4); // 262144 float4
  if (tid < NV4) {
    const int d4 = tid & (DST / 4 - 1);        // 0..127
    const int s  = (tid >> 7) & (NSTATES - 1); // 0..63
    ((v4f*)out)[tid] = ((const v4f*)bank)[s * (DST / 4) + d4];
  }
  if (tid < B_DIM * NSTATES) {
    const int b = tid >> 6;
    const int s = tid & (NSTATES - 1);
    out[(size_t)B_DIM * NSTATES * DST + tid] = (ranks[s] < num_states[b]) ? 1.0f : 0.0f;
  }
}

extern "C" void kernel_launch(void* const* d_in, const int* in_sizes, int n_in,
                              void* d_out, int out_size, void* d_ws, size_t ws_size,
                              hipStream_t stream) {
  (void)in_sizes; (void)n_in; (void)out_size; (void)ws_size;
  const float* x    = (const float*)d_in[0];
  const float* W1   = (const float*)d_in[1];
  const float* b1   = (const float*)d_in[2];
  const float* W2   = (const float*)d_in[3];
  const float* b2   = (const float*)d_in[4];
  const float* W3   = (const float*)d_in[5];
  const float* b3   = (const float*)d_in[6];
  const float* Ws1  = (const float*)d_in[7];
  const float* bs1  = (const float*)d_in[8];
  const float* Ws2  = (const float*)d_in[9];
  const float* bs2  = (const float*)d_in[10];
  const float* bank = (const float*)d_in[11];
  // d_in[12] (temperature) intentionally unused: softmax/temp are monotonic,
  // the top-k mask depends only on logit ordering.

  float* out = (float*)d_out;
  float* ws  = (float*)d_ws;
  float* xpool = ws;                               // 32*1024
  float* h1    = xpool + B_DIM * DIN;              // 32*512
  float* h2    = h1 + B_DIM * DST;                 // 32*256
  float* hs    = h2 + B_DIM * (DST / 2);           // 64*256
  int*   nsts  = (int*)(hs + NSTATES * (DST / 2)); // 32
  int*   ranks = nsts + B_DIM;                     // 64

  // Reuse d_out (4 MB allocated_states region) as the pool partial buffer;
  // it is fully overwritten by output_kernel at the end of the stream.
  float* partial = out; // needs 32*32*1024 floats = 4 MB <= 32*64*512 floats

  pool_stage1<<<dim3(SCHUNK, B_DIM), 256, 0, stream>>>(x, partial);
  pool_stage2<<<dim3(DIN / 256, B_DIM), 256, 0, stream>>>(partial, xpool);

  // h1 = relu(xpool @ W1^T + b1)        M=32, N=512, K=1024
  wmma_gemm_relu<<<dim3(DST / 16, B_DIM / 16), 32, 0, stream>>>(xpool, W1, b1, h1, B_DIM, DST, DIN);
  // h2 = relu(h1 @ W2^T + b2)           M=32, N=256, K=512
  wmma_gemm_relu<<<dim3(DST / 32, B_DIM / 16), 32, 0, stream>>>(h1, W2, b2, h2, B_DIM, DST / 2, DST);
  // hs = relu(bank @ Ws1^T + bs1)       M=64, N=256, K=512
  wmma_gemm_relu<<<dim3(DST / 32, NSTATES / 16), 32, 0, stream>>>(bank, Ws1, bs1, hs, NSTATES, DST / 2, DST);

  heads_kernel<<<1, 64, 0, stream>>>(h2, W3, b3, hs, Ws2, bs2, nsts, ranks);

  const int nthreads = B_DIM * NSTATES * (DST / 4); // 262144
  output_kernel<<<nthreads / 256, 256, 0, stream>>>(bank, nsts, ranks, out);
}